// DeformConv2d_82901458747486
// MI455X (gfx1250) — compile-verified
//
#include <hip/hip_runtime.h>

typedef float v2f __attribute__((ext_vector_type(2)));
typedef float v8f __attribute__((ext_vector_type(8)));

#define C_IN   256
#define C_OUT  256
#define Hd     64
#define Wd     64
#define Bd     4
#define Kd     9
#define KKTOT  (C_IN * Kd)      // 2304
#define NPIX   (Bd * Hd * Wd)   // 16384

// ---------------- kernel 0a: NCHW -> NHWC transpose of x ----------------
__global__ __launch_bounds__(256) void xpose_kernel(const float* __restrict__ x,
                                                    float* __restrict__ x_t) {
  __shared__ float tile[32][33];
  const int tx = threadIdx.x & 31;
  const int ty = threadIdx.x >> 5;            // 0..7
  const int bh = blockIdx.z;                  // b*64 + y
  const int b  = bh >> 6, y = bh & 63;
  const int c0 = blockIdx.y << 5;
  const int x0 = blockIdx.x << 5;
#pragma unroll
  for (int i = 0; i < 4; ++i) {
    int c = c0 + ty + i * 8;
    tile[ty + i * 8][tx] = x[(((b << 8) + c) * 64 + y) * 64 + x0 + tx]; // coalesced in x
  }
  __syncthreads();
#pragma unroll
  for (int i = 0; i < 4; ++i) {
    int w = x0 + ty + i * 8;
    x_t[((((b << 6) + y) * 64 + w) << 8) + c0 + tx] = tile[tx][ty + i * 8]; // coalesced in c
  }
}

// ------------- kernel 0b: pack weights into WMMA-friendly order -------------
// w_packed[o][k*256 + c] = weight[o][c][k]      (kk order matches sampling order)
// ow_packed[oc][ki*3+kj][c] = offset_w[oc][c][ki][kj]
__global__ __launch_bounds__(256) void pack_kernel(const float* __restrict__ weight,
                                                   const float* __restrict__ offw,
                                                   float* __restrict__ w_packed,
                                                   float* __restrict__ ow_packed) {
  const int tid = blockIdx.x * 256 + threadIdx.x;
  if (tid < C_OUT * KKTOT) {
    const int o  = tid / KKTOT;
    const int kk = tid - o * KKTOT;
    const int k  = kk >> 8;                 // kk = k*256 + c
    const int c  = kk & 255;
    w_packed[tid] = weight[(o * C_IN + c) * 9 + k];
  }
  if (tid < 18 * 9 * 256) {
    const int c    = tid & 255;
    const int rest = tid >> 8;              // oc*9 + kidx
    const int kidx = rest % 9;
    const int oc   = rest / 9;
    ow_packed[tid] = offw[(oc * C_IN + c) * 9 + kidx];
  }
}

// ---------------- kernel 1: offset-field conv (18 channels) ----------------
// One wave32 per output element; lanes = channels (coalesced NHWC reads).
__global__ __launch_bounds__(256) void offconv_kernel(const float* __restrict__ x_t,
                                                      const float* __restrict__ ow_packed,
                                                      const float* __restrict__ offb,
                                                      float* __restrict__ offs) {
  const int lane = threadIdx.x & 31;
  const int wv   = threadIdx.x >> 5;
  const int gid  = blockIdx.x * 8 + wv;     // = (b*18 + oc)*4096 + sp
  const int b    = gid / (18 * 4096);
  const int rem  = gid - b * 18 * 4096;
  const int oc   = rem >> 12;
  const int sp   = rem & 4095;
  const int oh = sp >> 6, ow = sp & 63;
  float sum = 0.f;
#pragma unroll
  for (int ki = 0; ki < 3; ++ki) {
    const int y = oh - 1 + ki;
    if ((unsigned)y >= 64u) continue;
#pragma unroll
    for (int kj = 0; kj < 3; ++kj) {
      const int x = ow - 1 + kj;
      if ((unsigned)x >= 64u) continue;
      const float* xp = x_t + (((((b << 6) + y) << 6) + x) << 8);
      const float* wp = ow_packed + ((oc * 9 + ki * 3 + kj) << 8);
#pragma unroll
      for (int cc = 0; cc < 8; ++cc) {
        const int c = (cc << 5) + lane;
        sum += xp[c] * wp[c];
      }
    }
  }
#pragma unroll
  for (int off = 16; off > 0; off >>= 1)
    sum += __shfl_xor(sum, off, 32);
  if (lane == 0) offs[gid] = sum + offb[oc];
}

// -------- kernel 2: bilinear sampling + WMMA fp32 GEMM (M=256,N=32/WG,K=2304) --------
__global__ __launch_bounds__(256, 1) void deform_main_kernel(
    const float* __restrict__ x_t, const float* __restrict__ w_packed,
    const float* __restrict__ offs, const float* __restrict__ bias,
    float* __restrict__ out) {
  __shared__ float smp[256 * 33];           // [c][pixel], padded stride 33 (33 KB)
  const int tid = threadIdx.x;
  const int pixBase = blockIdx.x << 5;      // 32 pixels per WG, never crosses batch
  const int b   = pixBase >> 12;
  const int sp0 = pixBase & 4095;
  const int wv = tid >> 5, lane = tid & 31;
  const int mBase = (wv >> 1) << 6;         // 4 M-groups of 64 rows
  const int nBase = (wv & 1) << 4;          // 2 N-groups of 16 pixels
  const bool hi  = lane >= 16;
  const int  l16 = lane & 15;

  v8f acc[4] = {v8f{}, v8f{}, v8f{}, v8f{}};

  const float* xb = x_t + ((long)b << 20);  // b * 64*64*256

  for (int k = 0; k < 9; ++k) {
    const int ki = k / 3, kj = k % 3;
    // ---- sampling: lane = channel (tid), coalesced 1KB taps from NHWC x ----
    for (int p = 0; p < 32; ++p) {
      const int sp = sp0 + p;
      const int oh = sp >> 6, ow = sp & 63;
      const float offy = offs[((b * 18 + 2 * k) << 12) + sp];
      const float offx = offs[((b * 18 + 2 * k + 1) << 12) + sp];
      const float py = (float)(oh - 1 + ki) + offy;
      const float px = (float)(ow - 1 + kj) + offx;
      const float y0f = floorf(py), x0f = floorf(px);
      const float wy = py - y0f, wx = px - x0f;
      const int y0 = (int)y0f, x0 = (int)x0f;
      const float w00 = (1.f - wy) * (1.f - wx);
      const float w01 = (1.f - wy) * wx;
      const float w10 = wy * (1.f - wx);
      const float w11 = wy * wx;
      float v = 0.f;
      if ((unsigned)y0 < 64u) {
        const float* row = xb + ((y0 << 6) << 8);
        if ((unsigned)x0 < 64u)       v += w00 * row[(x0 << 8) + tid];
        if ((unsigned)(x0 + 1) < 64u) v += w01 * row[((x0 + 1) << 8) + tid];
      }
      if ((unsigned)(y0 + 1) < 64u) {
        const float* row = xb + (((y0 + 1) << 6) << 8);
        if ((unsigned)x0 < 64u)       v += w10 * row[(x0 << 8) + tid];
        if ((unsigned)(x0 + 1) < 64u) v += w11 * row[((x0 + 1) << 8) + tid];
      }
      smp[tid * 33 + p] = v;                // 33*c mod 64 distinct -> conflict-free
    }
    __syncthreads();
    // ---- WMMA over this 256-deep K chunk: 64 steps x 4 M-tiles ----
    const int kg = k << 8;
    const float* wbase = w_packed + (long)(mBase + l16) * KKTOT + kg + (hi ? 2 : 0);
    for (int kc = 0; kc < 64; ++kc) {
      const int kb = (kc << 2) + (hi ? 2 : 0);   // A/B layout: lanes<16 -> K0,K1; lanes>=16 -> K2,K3
      v2f bv;
      bv.x = smp[kb * 33 + nBase + l16];
      bv.y = smp[(kb + 1) * 33 + nBase + l16];
#pragma unroll
      for (int t = 0; t < 4; ++t) {
        const v2f av = *(const v2f*)(wbase + (long)t * 16 * KKTOT + (kc << 2));
        acc[t] = __builtin_amdgcn_wmma_f32_16x16x4_f32(
            false, av, false, bv, (short)0, acc[t], false, false);
      }
    }
    __syncthreads();
  }
  // ---- epilogue: bias + store (C layout: VGPR r -> M=r / M=r+8) ----
#pragma unroll
  for (int t = 0; t < 4; ++t) {
#pragma unroll
    for (int r = 0; r < 8; ++r) {
      const int m = mBase + (t << 4) + (hi ? r + 8 : r);
      out[(((b << 8) + m) << 12) + sp0 + nBase + l16] = acc[t][r] + bias[m];
    }
  }
}

extern "C" void kernel_launch(void* const* d_in, const int* in_sizes, int n_in,
                              void* d_out, int out_size, void* d_ws, size_t ws_size,
                              hipStream_t stream) {
  const float* x      = (const float*)d_in[0];
  const float* weight = (const float*)d_in[1];
  const float* bias   = (const float*)d_in[2];
  const float* offw   = (const float*)d_in[3];
  const float* offb   = (const float*)d_in[4];
  float* out = (float*)d_out;

  float* ws        = (float*)d_ws;
  float* x_t       = ws;                    // 4,194,304 floats (16.8 MB)
  float* w_packed  = x_t + 4194304;         //   589,824 floats ( 2.4 MB)
  float* ow_packed = w_packed + 589824;     //    41,472 floats
  float* offs      = ow_packed + 41472;     //   294,912 floats ( 1.2 MB)

  xpose_kernel<<<dim3(2, 8, 256), 256, 0, stream>>>(x, x_t);
  pack_kernel<<<(C_OUT * KKTOT + 255) / 256, 256, 0, stream>>>(weight, offw, w_packed, ow_packed);
  offconv_kernel<<<(Bd * 18 * 4096) / 8, 256, 0, stream>>>(x_t, ow_packed, offb, offs);
  deform_main_kernel<<<NPIX / 32, 256, 0, stream>>>(x_t, w_packed, offs, bias, out);
}